// WN_s_51238959841344
// MI455X (gfx1250) — compile-verified
//
#include <hip/hip_runtime.h>

typedef __attribute__((ext_vector_type(2))) float v2f;
typedef __attribute__((ext_vector_type(8))) float v8f;

#define Bsz 32
#define Hsz 56
#define Wsz 56
#define Csz 256
#define TW 16
#define LDS_COLS 18     // TW + 2 halo columns
#define CH_PAD 260      // channel stride padding -> conflict-free pixel-strided LDS reads
#define TILES_W 4       // ceil(56/16)

__global__ __launch_bounds__(256) void gdn_fused_wmma_kernel(
    const float* __restrict__ x,          // (B,H,W,C)
    const float* __restrict__ gk,         // (8,C)   gamma_k[0,0]
    const float* __restrict__ gs,         // (9,C)   gamma_s[dy,dx,c,0] flattened
    const float* __restrict__ beta,       // (C,)
    const float* __restrict__ beta_out,   // (C,)
    const float* __restrict__ gamma_out,  // (C,)
    float* __restrict__ out)              // (B,H,W,C)
{
    __shared__ float lds[3 * LDS_COLS * CH_PAD];   // 56160 B, holds raw x (halo zero-padded)

    const int tid = threadIdx.x;
    int blk = blockIdx.x;
    const int tw = blk % TILES_W; blk /= TILES_W;
    const int h  = blk % Hsz;
    const int b  = blk / Hsz;
    const int w0 = tw * TW;

    // ---- cooperative load: 3 rows x 18 cols x 64 float4 of x into LDS ----
    for (int idx = tid; idx < 3 * LDS_COLS * (Csz / 4); idx += 256) {
        const int c4  = idx & 63;
        const int col = (idx >> 6) % LDS_COLS;
        const int row = idx / (64 * LDS_COLS);
        const int hh = h + row - 1;
        const int ww = w0 + col - 1;
        float4 v = make_float4(0.f, 0.f, 0.f, 0.f);
        if (hh >= 0 && hh < Hsz && ww >= 0 && ww < Wsz) {
            v = *((const float4*)(x + (((size_t)b * Hsz + hh) * Wsz + ww) * Csz) + c4);
        }
        float* d = lds + (row * LDS_COLS + col) * CH_PAD + c4 * 4;
        d[0] = v.x; d[1] = v.y; d[2] = v.z; d[3] = v.w;
    }
    __syncthreads();

    const int lane  = tid & 31;
    const int wv    = tid >> 5;
    const int nloc  = lane & 15;         // channel-in-tile (B/C/D), pixel row (A)
    const int khalf = (lane >> 4) * 2;   // upper half-wave holds K=k+2,k+3
    const int poff  = (lane >> 4) * 8;   // C/D: lanes 16-31 hold pixels M=8..15

    #pragma unroll
    for (int t = 0; t < 2; ++t) {
        const int ct = wv * 32 + t * 16;   // channel tile base for this wave

        // ---- B fragments: block-diagonal gamma_k tile (two 8x8 blocks) ----
        v2f bfrag[4];
        #pragma unroll
        for (int s = 0; s < 4; ++s) {
            const int k = s * 4 + khalf;
            float b0 = 0.f, b1 = 0.f;
            if ((k >> 3) == (nloc >> 3))       b0 = gk[(k & 7) * Csz + ct + nloc];
            if (((k + 1) >> 3) == (nloc >> 3)) b1 = gk[((k + 1) & 7) * Csz + ct + nloc];
            v2f bf; bf[0] = b0; bf[1] = b1;
            bfrag[s] = bf;
        }

        // ---- A fragments: x^2 of center pixels, accumulate Pk via f32 WMMA ----
        const float* cen = lds + (1 * LDS_COLS + 1 + nloc) * CH_PAD + ct;  // pixel = nloc
        v8f acc = {};
        #pragma unroll
        for (int s = 0; s < 4; ++s) {
            const int k = s * 4 + khalf;
            const float a0 = cen[k];
            const float a1 = cen[k + 1];
            v2f af; af[0] = a0 * a0; af[1] = a1 * a1;
            acc = __builtin_amdgcn_wmma_f32_16x16x4_f32(
                false, af, false, bfrag[s], (short)0, acc, false, false);
        }

        // ---- epilogue: Ps (8-tap depthwise), rsqrt, scale/shift, store ----
        const int ch = ct + nloc;
        float ks[8];
        {
            int ti = 0;
            #pragma unroll
            for (int tap = 0; tap < 9; ++tap) {
                if (tap == 4) continue;              // center masked to zero
                ks[ti++] = gs[tap * Csz + ch];
            }
        }
        const float bv = beta[ch] + 1e-6f;
        const float go = gamma_out[ch];
        const float bo = beta_out[ch];

        #pragma unroll
        for (int j = 0; j < 8; ++j) {
            const int px = j + poff;                 // pixel within tile
            const int wg = w0 + px;
            if (wg >= Wsz) continue;                 // last tile partially OOB
            float ps = 0.f;
            int ti = 0;
            #pragma unroll
            for (int dy = 0; dy < 3; ++dy) {
                #pragma unroll
                for (int dx = 0; dx < 3; ++dx) {
                    if (dy == 1 && dx == 1) continue;
                    const float xv = lds[(dy * LDS_COLS + px + dx) * CH_PAD + ch];
                    ps = fmaf(ks[ti], xv * xv, ps);
                    ++ti;
                }
            }
            const float xc = lds[(1 * LDS_COLS + 1 + px) * CH_PAD + ch];
            const float pk = acc[j];
            out[(((size_t)b * Hsz + h) * Wsz + wg) * Csz + ch] =
                xc * rsqrtf(pk + ps + bv) * go + bo;
        }
    }
}

extern "C" void kernel_launch(void* const* d_in, const int* in_sizes, int n_in,
                              void* d_out, int out_size, void* d_ws, size_t ws_size,
                              hipStream_t stream) {
    const float* x    = (const float*)d_in[0];
    const float* gk   = (const float*)d_in[1];
    const float* gs   = (const float*)d_in[2];
    const float* beta = (const float*)d_in[3];
    const float* bo   = (const float*)d_in[4];
    const float* go   = (const float*)d_in[5];
    // d_in[6] = surround_dist == 1 (hardcoded in the 3x3 halo layout)

    dim3 grid(Bsz * Hsz * TILES_W);   // 7168 blocks
    gdn_fused_wmma_kernel<<<grid, 256, 0, stream>>>(x, gk, gs, beta, bo, go, (float*)d_out);
}